// GCNLayer_31421980738083
// MI455X (gfx1250) — compile-verified
//
#include <hip/hip_runtime.h>
#include <stdint.h>

typedef __attribute__((ext_vector_type(2))) float v2f;
typedef __attribute__((ext_vector_type(8))) float v8f;
typedef __attribute__((ext_vector_type(4))) unsigned int v4u;
typedef __attribute__((ext_vector_type(8))) unsigned int v8u;

#define N_ROWS    16384
#define IN_DIM    256
#define OUT_DIM   128
#define KCHUNK    32
#define ROWSTRIDE 136   // 128 data dwords + 8 pad dwords inserted by TDM per row

// ---------------------------------------------------------------------------
// Kernel 1: h = x @ W^T + b     [16384,256] x [256,128] -> [16384,128]
// 128 blocks x 256 threads (8 waves). Wave w computes rows m0..m0+15, all 128
// cols via 8 accumulator tiles of V_WMMA_F32_16X16X4_F32.
// ---------------------------------------------------------------------------
__global__ __launch_bounds__(256)
void dense_xwT_bias(const float* __restrict__ x, const float* __restrict__ W,
                    const float* __restrict__ bias, float* __restrict__ h) {
  const int lane = threadIdx.x & 31;
  const int wave = threadIdx.x >> 5;
  const int lcol = lane & 15;   // M (for A) / N (for B,C) index within tile
  const int half = lane >> 4;   // K-pair selector for A/B fragments
  const int m0   = blockIdx.x * 128 + wave * 16;

  v8f acc[8];
#pragma unroll
  for (int j = 0; j < 8; ++j) acc[j] = (v8f)0.0f;

  // A fragment: lane<16 -> x[m0+lcol][k..k+1], lane>=16 -> x[m0+lcol][k+2..k+3]
  const float* arow = x + (size_t)(m0 + lcol) * IN_DIM + 2 * half;

  for (int k = 0; k < IN_DIM; k += 4) {
    v2f a = *(const v2f*)(arow + k);
#pragma unroll
    for (int j = 0; j < 8; ++j) {
      // B[k][n] = W[n][k]; lane<16 holds rows k,k+1 at col n0+lcol (contiguous in W)
      const float* bp = W + (size_t)(j * 16 + lcol) * IN_DIM + k + 2 * half;
      v2f bf = *(const v2f*)bp;
      acc[j] = __builtin_amdgcn_wmma_f32_16x16x4_f32(
          false, a, false, bf, (short)0, acc[j], false, false);
    }
  }

  // C/D layout: VGPR r -> M = r (+8 for upper lane half), N = lcol
#pragma unroll
  for (int j = 0; j < 8; ++j) {
    const int n = j * 16 + lcol;
    const float bv = bias[n];
#pragma unroll
    for (int r = 0; r < 8; ++r) {
      const int m = m0 + r + 8 * half;
      h[(size_t)m * OUT_DIM + n] = acc[j][r] + bv;
    }
  }
}

// ---------------------------------------------------------------------------
// TDM issue: async-copy h[kbase..kbase+31][0..127] (row-major fp32) into LDS
// at byte offset lds_off, inserting 8 pad dwords after every 128 data dwords
// (-> 136-dword row stride, bank-conflict-free fragment reads).
// 2D descriptor: groups 0 and 1 only (VADDR2/VADDR3 omitted -> NULL).
// ---------------------------------------------------------------------------
__device__ __forceinline__ void tdm_load_chunk(const float* gsrc, unsigned lds_off) {
  const unsigned long long ga = (unsigned long long)(uintptr_t)gsrc;
  union { unsigned w[4]; v4u v; } g0;
  union { unsigned w[8]; v8u v; } g1;

  g0.w[0] = 1u;                                     // count=1, user D#, no gather
  g0.w[1] = lds_off;                                // lds_addr (bytes)
  g0.w[2] = (unsigned)(ga & 0xFFFFFFFFu);           // global_addr[31:0]
  g0.w[3] = (unsigned)((ga >> 32) & 0x01FFFFFFu)    // global_addr[56:32]
          | (2u << 30);                             // type = 2 ("image")

  g1.w[0] = (2u << 16)    // data_size = 4 bytes
          | (1u << 20)    // pad_enable
          | (6u << 22)    // pad_interval: pad after 128 dwords
          | (7u << 25);   // pad_amount: 8 dwords
  g1.w[1] = ((unsigned)OUT_DIM & 0xFFFFu) << 16;    // tensor_dim0[15:0] = 128
  g1.w[2] = (((unsigned)N_ROWS) & 0xFFFFu) << 16;   // dim0 hi=0 | tensor_dim1 lo16
  g1.w[3] = (((unsigned)N_ROWS) >> 16)              // tensor_dim1 hi16 (=0)
          | ((unsigned)OUT_DIM << 16);              // tile_dim0 = 128
  g1.w[4] = (unsigned)KCHUNK;                       // tile_dim1 = 32, tile_dim2 = 0
  g1.w[5] = (unsigned)OUT_DIM;                      // tensor_dim0_stride = 128
  g1.w[6] = 0u;                                     // stride0 hi | dim1_stride lo
  g1.w[7] = 0u;

  asm volatile("tensor_load_to_lds %0, %1" :: "s"(g0.v), "s"(g1.v) : "memory");
}

// ---------------------------------------------------------------------------
// Kernel 2: out = A_hat @ h     [16384,16384] x [16384,128] -> [16384,128]
// HBM-bound on A_hat (1 GiB fp32, read exactly once). h staged per-block into
// LDS by the Tensor Data Mover, double-buffered: DMA of chunk i+1 overlaps the
// WMMA crunch of chunk i. Wave 0 issues the TDM op and gates completion with
// s_wait_tensorcnt; the workgroup barrier publishes the buffer to all 8 waves.
// ---------------------------------------------------------------------------
__global__ __launch_bounds__(256)
void agg_spmm(const float* __restrict__ A, const float* __restrict__ h,
              float* __restrict__ out) {
  __shared__ __align__(16) float Bs[2][KCHUNK][ROWSTRIDE];

  const int tid  = threadIdx.x;
  const int lane = tid & 31;
  const int wave = tid >> 5;
  const int lcol = lane & 15;
  const int half = lane >> 4;
  const int m0   = blockIdx.x * 128 + wave * 16;

  v8f acc[8];
#pragma unroll
  for (int j = 0; j < 8; ++j) acc[j] = (v8f)0.0f;

  const unsigned lds_base  = (unsigned)(uintptr_t)(&Bs[0][0][0]);
  const unsigned buf_bytes = KCHUNK * ROWSTRIDE * 4;

  const float* arow = A + (size_t)(m0 + lcol) * N_ROWS + 2 * half;

  // Prologue: kick off DMA of chunk 0 into buffer 0.
  if (wave == 0) tdm_load_chunk(h, lds_base);

  int ibuf = 0;
  for (int kbase = 0; kbase < N_ROWS; kbase += KCHUNK, ibuf ^= 1) {
    const bool has_next = (kbase + KCHUNK) < N_ROWS;
    if (wave == 0) {
      if (has_next)
        tdm_load_chunk(h + (size_t)(kbase + KCHUNK) * OUT_DIM,
                       lds_base + (unsigned)(ibuf ^ 1) * buf_bytes);
      // TDM ops complete in order: cnt<=1 means chunk `kbase` has landed.
      if (has_next) __builtin_amdgcn_s_wait_tensorcnt(1);
      else          __builtin_amdgcn_s_wait_tensorcnt(0);
    }
    __syncthreads();

    // Prefetch this wave's next A chunk while crunching (global_prefetch_b8).
    if (has_next) __builtin_prefetch(arow + kbase + KCHUNK, 0, 0);

#pragma unroll
    for (int k = 0; k < KCHUNK; k += 4) {
      v2f a = *(const v2f*)(arow + kbase + k);
      const int r0 = k + 2 * half;   // B rows for this lane half: r0, r0+1
#pragma unroll
      for (int j = 0; j < 8; ++j) {
        const int n = j * 16 + lcol;
        v2f bf;
        bf.x = Bs[ibuf][r0][n];
        bf.y = Bs[ibuf][r0 + 1][n];
        acc[j] = __builtin_amdgcn_wmma_f32_16x16x4_f32(
            false, a, false, bf, (short)0, acc[j], false, false);
      }
    }
    __syncthreads();   // nobody may still read buf (ibuf^1) when DMA reuses it
  }

#pragma unroll
  for (int j = 0; j < 8; ++j) {
    const int n = j * 16 + lcol;
#pragma unroll
    for (int r = 0; r < 8; ++r) {
      const int m = m0 + r + 8 * half;
      out[(size_t)m * OUT_DIM + n] = acc[j][r];
    }
  }
}

// ---------------------------------------------------------------------------
extern "C" void kernel_launch(void* const* d_in, const int* in_sizes, int n_in,
                              void* d_out, int out_size, void* d_ws, size_t ws_size,
                              hipStream_t stream) {
  const float* x     = (const float*)d_in[0];  // [16384, 256]
  const float* A_hat = (const float*)d_in[1];  // [16384, 16384]
  const float* W     = (const float*)d_in[2];  // [128, 256]
  const float* b     = (const float*)d_in[3];  // [128]
  float* out = (float*)d_out;                  // [16384, 128]
  float* h   = (float*)d_ws;                   // [16384, 128] scratch (8 MB)

  dense_xwT_bias<<<N_ROWS / 128, 256, 0, stream>>>(x, W, b, h);
  agg_spmm<<<N_ROWS / 128, 256, 0, stream>>>(A_hat, h, out);
}